// SpDHRGATLayer_84954453115142
// MI455X (gfx1250) — compile-verified
//
#include <hip/hip_runtime.h>
#include <hip/hip_bf16.h>
#include <math.h>

typedef __attribute__((ext_vector_type(2))) float v2f;
typedef __attribute__((ext_vector_type(8))) float v8f;

#define IN_F   128
#define OUT_F  128
#define HID_F  512
#define ALPHA  0.2f

// ---------------------------------------------------------------------------
// K_zero: zero accumulators (d_out used as h_accum, rowsum in ws)
// ---------------------------------------------------------------------------
__global__ void k_zero(float* __restrict__ out, float* __restrict__ rowsum,
                       int n_out, int n_nodes) {
    int i = blockIdx.x * blockDim.x + threadIdx.x;
    if (i < n_out)   out[i]    = 0.0f;
    if (i < n_nodes) rowsum[i] = 0.0f;
}

// ---------------------------------------------------------------------------
// K_prep: wa1 = W@a1, wa3 = W@a3, wr_a2 = W_r@a2, c_dot = c_r . wr_a2
// single block, 128 threads
// ---------------------------------------------------------------------------
__global__ void k_prep(const float* __restrict__ W, const float* __restrict__ W_r,
                       const float* __restrict__ a, const float* __restrict__ c_r,
                       float* __restrict__ wa1, float* __restrict__ wa3,
                       float* __restrict__ wr_a2, float* __restrict__ c_dot) {
    __shared__ float red[IN_F];
    int k = threadIdx.x;                      // 0..127 (row of W / W_r)
    float v1 = 0.f, v3 = 0.f, v2 = 0.f;
    for (int j = 0; j < OUT_F; ++j) {
        float wj  = W[k * OUT_F + j];
        float wrj = W_r[k * OUT_F + j];
        v1 += wj  * a[j];
        v3 += wj  * a[2 * OUT_F + j];
        v2 += wrj * a[OUT_F + j];
    }
    wa1[k] = v1; wa3[k] = v3; wr_a2[k] = v2;
    red[k] = c_r[k] * v2;
    __syncthreads();
    for (int s = 64; s > 0; s >>= 1) {
        if (k < s) red[k] += red[k + s];
        __syncthreads();
    }
    if (k == 0) c_dot[0] = red[0];
}

// ---------------------------------------------------------------------------
// K_theta: per-relation MLP -> g_table[r][k] = cos(theta_r[k]) * wr_a2[k]
// one block per relation, 128 threads
// ---------------------------------------------------------------------------
__global__ void k_theta(const float* __restrict__ re, const float* __restrict__ w1,
                        const float* __restrict__ b1, const float* __restrict__ w2,
                        const float* __restrict__ b2, const float* __restrict__ wr_a2,
                        float* __restrict__ g_table) {
    __shared__ float hid[HID_F];
    __shared__ float rrow[IN_F];
    int r = blockIdx.x;
    int t = threadIdx.x;                      // 0..127
    rrow[t] = re[r * IN_F + t];
    __syncthreads();
    for (int h = t; h < HID_F; h += IN_F) {
        float acc = b1[h];
        for (int k = 0; k < IN_F; ++k) acc += rrow[k] * w1[k * HID_F + h];
        hid[h] = acc > 0.f ? acc : 0.f;
    }
    __syncthreads();
    float th = b2[t];
    for (int h = 0; h < HID_F; ++h) th += hid[h] * w2[h * IN_F + t];
    g_table[r * IN_F + t] = cosf(th) * wr_a2[t];
}

// ---------------------------------------------------------------------------
// K_gemm: Wh = x @ W via V_WMMA_F32_16X16X4_F32.
// block = 256 threads (8 waves). W staged in 64KB LDS. Each wave computes one
// 16-row strip x all 128 output columns (8 16x16 tiles, K-loop of 32 WMMAs).
// nrows (50000) is a multiple of 16, so every strip is full and EXEC stays
// all-ones inside the WMMA region (wave-uniform early exit only).
// ---------------------------------------------------------------------------
__global__ __launch_bounds__(256) void k_gemm(const float* __restrict__ x,
                                              const float* __restrict__ W,
                                              float* __restrict__ Wh, int nstrips) {
    __shared__ float sW[IN_F * OUT_F];        // 64 KB
    for (int i = threadIdx.x; i < IN_F * OUT_F; i += 256) sW[i] = W[i];
    __syncthreads();

    int wave = threadIdx.x >> 5;
    int lane = threadIdx.x & 31;
    int strip = blockIdx.x * 8 + wave;
    if (strip >= nstrips) return;             // uniform per wave

    int row_base = strip * 16;
    int r        = lane & 15;                 // M row / N col within tile
    int koff     = (lane >> 4) * 2;           // lanes 16..31 hold K+2,K+3

    v8f acc[8];
#pragma unroll
    for (int t = 0; t < 8; ++t) acc[t] = (v8f){0.f,0.f,0.f,0.f,0.f,0.f,0.f,0.f};

    const float* xrow = x + (size_t)(row_base + r) * IN_F;
    for (int k0 = 0; k0 < IN_F; k0 += 4) {
        int ka = k0 + koff;
        v2f av;
        av.x = xrow[ka];
        av.y = xrow[ka + 1];
#pragma unroll
        for (int t = 0; t < 8; ++t) {
            int n0 = t * 16;
            v2f bv;
            bv.x = sW[ka * OUT_F + n0 + r];
            bv.y = sW[(ka + 1) * OUT_F + n0 + r];
            acc[t] = __builtin_amdgcn_wmma_f32_16x16x4_f32(
                false, av, false, bv, (short)0, acc[t], false, false);
        }
    }
#pragma unroll
    for (int t = 0; t < 8; ++t) {
        int n = t * 16 + r;
#pragma unroll
        for (int v = 0; v < 8; ++v) {
            int m = row_base + v + ((lane < 16) ? 0 : 8);
            Wh[(size_t)m * OUT_F + n] = acc[t][v];
        }
    }
}

// ---------------------------------------------------------------------------
// K_node: s1[n] = x[n].wa1 ; s3[n] = x[n].wa3   (one wave per node)
// ---------------------------------------------------------------------------
__global__ __launch_bounds__(256) void k_node(const float* __restrict__ x,
                                              const float* __restrict__ wa1,
                                              const float* __restrict__ wa3,
                                              float* __restrict__ s1,
                                              float* __restrict__ s3, int n) {
    __shared__ float4 sa1[32], sa3[32];
    if (threadIdx.x < 32) {
        sa1[threadIdx.x] = ((const float4*)wa1)[threadIdx.x];
        sa3[threadIdx.x] = ((const float4*)wa3)[threadIdx.x];
    }
    __syncthreads();
    int lane = threadIdx.x & 31;
    int node = blockIdx.x * 8 + (threadIdx.x >> 5);
    if (node >= n) return;
    float4 xv = ((const float4*)(x + (size_t)node * IN_F))[lane];
    float4 a1v = sa1[lane], a3v = sa3[lane];
    float p1 = xv.x*a1v.x + xv.y*a1v.y + xv.z*a1v.z + xv.w*a1v.w;
    float p3 = xv.x*a3v.x + xv.y*a3v.y + xv.z*a3v.z + xv.w*a3v.w;
    for (int off = 16; off > 0; off >>= 1) {
        p1 += __shfl_xor(p1, off, 32);
        p3 += __shfl_xor(p3, off, 32);
    }
    if (lane == 0) { s1[node] = p1; s3[node] = p3; }
}

// ---------------------------------------------------------------------------
// K_edge: per-edge attention weight + atomic scatter (one wave per edge)
// score = s1[src] + s3[dst] + c_dot + x[src].g[type]
// edge_e = exp(-lrelu(score)); hacc[src] += edge_e * Wh[dst]; rowsum[src] += e
// ---------------------------------------------------------------------------
__global__ __launch_bounds__(256) void k_edge(const float* __restrict__ x,
                                              const int* __restrict__ src_a,
                                              const int* __restrict__ dst_a,
                                              const int* __restrict__ etype,
                                              const float* __restrict__ g_table,
                                              const float* __restrict__ s1,
                                              const float* __restrict__ s3,
                                              const float* __restrict__ Wh,
                                              const float* __restrict__ c_dot,
                                              float* __restrict__ rowsum,
                                              float* __restrict__ hacc, int E) {
    int lane = threadIdx.x & 31;
    int e = blockIdx.x * 8 + (threadIdx.x >> 5);
    if (e >= E) return;                       // uniform per wave
    int s = src_a[e], d = dst_a[e], t = etype[e];

    float4 xv = ((const float4*)(x + (size_t)s * IN_F))[lane];
    float4 gv = ((const float4*)(g_table + (size_t)t * IN_F))[lane];
    float p = xv.x*gv.x + xv.y*gv.y + xv.z*gv.z + xv.w*gv.w;
    for (int off = 16; off > 0; off >>= 1) p += __shfl_xor(p, off, 32);

    float score = s1[s] + s3[d] + c_dot[0] + p;
    float lr = score > 0.f ? score : ALPHA * score;
    float w = __expf(-lr);

    if (lane == 0) atomicAdd(&rowsum[s], w);

    float4 wv = ((const float4*)(Wh + (size_t)d * OUT_F))[lane];
    float* outp = hacc + (size_t)s * OUT_F + lane * 4;
    atomicAdd(outp + 0, w * wv.x);
    atomicAdd(outp + 1, w * wv.y);
    atomicAdd(outp + 2, w * wv.z);
    atomicAdd(outp + 3, w * wv.w);
}

// ---------------------------------------------------------------------------
// K_final: h = hacc / rowsum (guard 0 -> 1e-12), then ELU, in-place in d_out
// ---------------------------------------------------------------------------
__global__ void k_final(float* __restrict__ out, const float* __restrict__ rowsum,
                        int n_elems) {
    int i = blockIdx.x * blockDim.x + threadIdx.x;
    if (i >= n_elems) return;
    float rs = rowsum[i >> 7];
    if (rs == 0.f) rs = 1e-12f;
    float h = out[i] / rs;
    out[i] = h > 0.f ? h : expm1f(h);
}

// ---------------------------------------------------------------------------
extern "C" void kernel_launch(void* const* d_in, const int* in_sizes, int n_in,
                              void* d_out, int out_size, void* d_ws, size_t ws_size,
                              hipStream_t stream) {
    const float* x    = (const float*)d_in[0];
    const float* re   = (const float*)d_in[1];
    const float* w1   = (const float*)d_in[2];
    const float* b1   = (const float*)d_in[3];
    const float* w2   = (const float*)d_in[4];
    const float* b2   = (const float*)d_in[5];
    const float* c_r  = (const float*)d_in[6];
    const float* W    = (const float*)d_in[7];
    const float* W_r  = (const float*)d_in[8];
    const float* a    = (const float*)d_in[9];
    const int*   el   = (const int*)d_in[10];   // [2, E]
    const int*   et   = (const int*)d_in[11];   // [E]

    const int N = in_sizes[0] / IN_F;
    const int R = in_sizes[1] / IN_F;
    const int E = in_sizes[11];
    const int* src_a = el;
    const int* dst_a = el + E;

    float* out = (float*)d_out;                 // [N, OUT] also used as h accum
    float* ws  = (float*)d_ws;

    // workspace layout (floats, all float4-aligned where needed)
    float* g_table = ws;                        // R*128
    float* wa1     = g_table + (size_t)R * IN_F;
    float* wa3     = wa1 + IN_F;
    float* wr_a2   = wa3 + IN_F;
    float* c_dot   = wr_a2 + IN_F;              // 1 (pad to 4)
    float* s1      = c_dot + 4;
    float* s3      = s1 + N;
    size_t whoff   = ((size_t)(s3 + N - ws) + 3) & ~(size_t)3;
    float* Wh      = ws + whoff;                // N*128
    float* rowsum  = Wh + (size_t)N * OUT_F;    // N

    const int n_out = N * OUT_F;

    // 1) zero accumulators (required every call: graph replays accumulate)
    k_zero<<<(n_out + 255) / 256, 256, 0, stream>>>(out, rowsum, n_out, N);

    // 2) fold attention vectors into weights
    k_prep<<<1, IN_F, 0, stream>>>(W, W_r, a, c_r, wa1, wa3, wr_a2, c_dot);

    // 3) per-relation theta table -> g_table
    k_theta<<<R, IN_F, 0, stream>>>(re, w1, b1, w2, b2, wr_a2, g_table);

    // 4) Wh = x @ W  (WMMA fp32)
    int nstrips = N / 16;                       // N = 50000 -> 3125 exact
    k_gemm<<<(nstrips + 7) / 8, 256, 0, stream>>>(x, W, Wh, nstrips);

    // 5) per-node score scalars
    k_node<<<(N + 7) / 8, 256, 0, stream>>>(x, wa1, wa3, s1, s3, N);

    // 6) per-edge attention + atomic scatter into out / rowsum
    k_edge<<<(E + 7) / 8, 256, 0, stream>>>(x, src_a, dst_a, et, g_table,
                                            s1, s3, Wh, c_dot, rowsum, out, E);

    // 7) normalize + ELU
    k_final<<<(n_out + 255) / 256, 256, 0, stream>>>(out, rowsum, n_out);
}